// SRP_grid_map_4200478015557
// MI455X (gfx1250) — compile-verified
//
#include <hip/hip_runtime.h>

// SRP grid-map gather-reduce for MI455X (gfx1250, wave32).
//   x:    (B*T=64, P=144, K=4096) f32
//   tau0: (P=144, G=8192) i32, values in [0, K)
//   out:  (64, 8192) f32  =  normalize( sum_p x[b, p, tau0[p,g]] )
//
// Pipeline: double-buffered global_load_async_to_lds_b128 staging (ASYNCcnt)
// overlapped with ds_load gathers + v_wmma_f32_16x16x4_f32 pair reduction.

#define BT      64
#define NP      144
#define KK      4096
#define GG      8192
#define GSPLIT  8
#define GS      (GG / GSPLIT)   // 1024 directions per block
#define PCHUNK  4               // x rows staged per step (4 * 16KB = 64KB)
#define NCHUNK  (NP / PCHUNK)   // 36 chunks
#define THREADS 256             // 8 wave32
#define NSTAGE  ((PCHUNK * KK * 4) / (THREADS * 16))   // 16 B128 DMAs / thread

typedef __attribute__((ext_vector_type(2))) float v2f;
typedef __attribute__((ext_vector_type(8))) float v8f;

typedef __attribute__((address_space(3))) void* lds_vptr_t;

// LDS byte offset of a generic pointer known to live in LDS.
__device__ __forceinline__ unsigned lds_addr_of(const void* p) {
  return (unsigned)(unsigned long long)(lds_vptr_t)(void*)p;
}

__global__ __launch_bounds__(THREADS)
void srp_sum_kernel(const float* __restrict__ x,
                    const int*   __restrict__ tau0,
                    float*       __restrict__ out) {
  __shared__ float smem[2][PCHUNK * KK];   // 2 x 64 KB double buffer

  const int b     = blockIdx.x >> 3;            // batch map 0..63
  const int gs    = blockIdx.x & (GSPLIT - 1);  // direction slice
  const int gbase = gs * GS;
  const int tid   = threadIdx.x;
  const int wave  = tid >> 5;
  const int lane  = tid & 31;
  const int nlow  = lane & 15;   // B-matrix column within the 16-wide g tile
  const int hi    = lane >> 4;   // lane half selects pair rows {0,1} vs {2,3}

  const float*   xb       = x + (size_t)b * NP * KK;
  const unsigned lds_base = lds_addr_of(&smem[0][0]);

  // Issue one chunk's async DMA: 16 back-to-back B128 transfers per thread,
  // single base address + literal INST_OFFSET (applies to LDS and global).
  auto stage = [&](int pc, int buf) {
    const char*    ga   = (const char*)(xb + (size_t)pc * KK) + (size_t)tid * 16;
    const unsigned loff = lds_base + (unsigned)(buf * (PCHUNK * KK * 4))
                                   + (unsigned)(tid * 16);
    #pragma unroll
    for (int j = 0; j < NSTAGE; ++j) {
      asm volatile("global_load_async_to_lds_b128 %0, %1, off offset:%c2"
                   :: "v"(loff), "v"(ga), "i"(j * (THREADS * 16)) : "memory");
    }
  };

  v8f acc[8];
  #pragma unroll
  for (int i = 0; i < 8; ++i) { v8f z = {}; acc[i] = z; }

  v2f ones; ones.x = 1.0f; ones.y = 1.0f;   // A = all-ones 16x4

  // Prime the pipeline with chunk 0.
  stage(0, 0);
  asm volatile("s_wait_asynccnt 0" ::: "memory");
  __syncthreads();

  for (int c = 0; c < NCHUNK; ++c) {
    const int pc = c * PCHUNK;

    // Kick off the next chunk's DMA before consuming the current buffer.
    if (c + 1 < NCHUNK) stage(pc + PCHUNK, (c + 1) & 1);

    const float* bufp = &smem[c & 1][0];

    // lanes 0-15 carry pairs (pc+0, pc+1); lanes 16-31 carry (pc+2, pc+3)
    const int  prow0 = hi << 1;
    const int* t0    = tau0 + (size_t)(pc + prow0) * GG;
    const int* t1    = t0 + GG;

    #pragma unroll
    for (int pass = 0; pass < 8; ++pass) {
      const int g  = gbase + pass * 128 + wave * 16 + nlow;
      const int k0 = t0[g];
      const int k1 = t1[g];
      v2f bm;
      bm.x = bufp[prow0 * KK + k0];          // ds_load gathers (64-bank LDS)
      bm.y = bufp[(prow0 + 1) * KK + k1];
      // D = ones(16x4) * gathered(4x16) + C : every row of D = column sums
      acc[pass] = __builtin_amdgcn_wmma_f32_16x16x4_f32(
          false, ones, false, bm, (short)0, acc[pass], false, false);
    }

    // Publish the next buffer: my async DMA done, then block-wide barrier.
    asm volatile("s_wait_asynccnt 0" ::: "memory");
    __syncthreads();
  }

  // Row M=0 of each D tile lives in VGPR0 of lanes 0-15 (N = lane).
  if (lane < 16) {
    #pragma unroll
    for (int pass = 0; pass < 8; ++pass) {
      const int g = gbase + pass * 128 + wave * 16 + lane;
      out[(size_t)b * GG + g] = acc[pass][0];
    }
  }
}

__global__ __launch_bounds__(THREADS)
void srp_norm_kernel(float* __restrict__ out) {
  __shared__ float red[THREADS];
  const int b   = blockIdx.x;
  const int tid = threadIdx.x;
  float* m = out + (size_t)b * GG;

  float v[GG / THREADS];                 // 32 values per thread
  float mx = -3.402823466e38f;
  #pragma unroll
  for (int i = 0; i < GG / THREADS; ++i) {
    float t = m[tid + i * THREADS] + 1e-12f;
    v[i] = t;
    mx = fmaxf(mx, t);
  }
  red[tid] = mx;
  __syncthreads();
  for (int s = THREADS / 2; s > 0; s >>= 1) {
    if (tid < s) red[tid] = fmaxf(red[tid], red[tid + s]);
    __syncthreads();
  }
  const float mmax = red[0];
  #pragma unroll
  for (int i = 0; i < GG / THREADS; ++i) {
    m[tid + i * THREADS] = v[i] / mmax;
  }
}

extern "C" void kernel_launch(void* const* d_in, const int* in_sizes, int n_in,
                              void* d_out, int out_size, void* d_ws, size_t ws_size,
                              hipStream_t stream) {
  (void)in_sizes; (void)n_in; (void)out_size; (void)d_ws; (void)ws_size;
  const float* x    = (const float*)d_in[0];
  const int*   tau0 = (const int*)d_in[1];
  float*       out  = (float*)d_out;

  srp_sum_kernel<<<dim3(BT * GSPLIT), dim3(THREADS), 0, stream>>>(x, tau0, out);
  srp_norm_kernel<<<dim3(BT), dim3(THREADS), 0, stream>>>(out);
}